// OTAM_Loss_12781822673484
// MI455X (gfx1250) — compile-verified
//
#include <hip/hip_runtime.h>
#include <math.h>

// Problem constants (from reference setup_inputs)
#define B_    8
#define NS_   5
#define NQ_   128
#define T_    16
#define D_    2048
#define LAM_  0.1f
#define INVLAM_ 10.0f
#define EPS_  1e-8f

#define N_MAT (B_*NQ_*NS_)                 // 5120 dist matrices
#define N_SUPP_ROWS (B_*NS_*T_)            // 640
#define N_QUERY_ROWS (B_*NQ_*T_)           // 16384

// Workspace layout (float offsets)
#define OFF_DIST 0
#define SZ_DIST  (N_MAT*T_*T_)             // 1,310,720
#define OFF_NORM (OFF_DIST + SZ_DIST)
#define SZ_NORM  (N_SUPP_ROWS + N_QUERY_ROWS)
#define OFF_TAM  (OFF_NORM + SZ_NORM)
#define SZ_TAM   (2*N_MAT)

typedef __attribute__((ext_vector_type(2))) float v2f;
typedef __attribute__((ext_vector_type(8))) float v8f;

#if defined(__has_builtin)
#  if __has_builtin(__builtin_amdgcn_wmma_f32_16x16x4_f32)
#    define HAVE_WMMA_F32X4 1
#  endif
#endif
#ifndef HAVE_WMMA_F32X4
#  define HAVE_WMMA_F32X4 0
#endif

// ---------------------------------------------------------------------------
// Kernel 1: row L2 norms.  One wave32 per row of 2048 f32.
// rows [0, 640) -> supp rows; rows [640, 17024) -> query rows.
// ---------------------------------------------------------------------------
__global__ __launch_bounds__(256) void otam_norms_kernel(
    const float* __restrict__ supp, const float* __restrict__ query,
    float* __restrict__ norm) {
  int warp = threadIdx.x >> 5;
  int lane = threadIdx.x & 31;
  int row = blockIdx.x * 8 + warp;
  const int NROWS = N_SUPP_ROWS + N_QUERY_ROWS;
  if (row >= NROWS) return;
  const float* p = (row < N_SUPP_ROWS)
                       ? (supp + (size_t)row * D_)
                       : (query + (size_t)(row - N_SUPP_ROWS) * D_);
  float s = 0.f;
#pragma unroll
  for (int i = 0; i < D_ / 128; ++i) {
    float4 v = *(const float4*)(p + i * 128 + lane * 4);
    s += v.x * v.x + v.y * v.y + v.z * v.z + v.w * v.w;
  }
#pragma unroll
  for (int m = 16; m >= 1; m >>= 1) s += __shfl_xor(s, m, 32);
  if (lane == 0) norm[row] = sqrtf(s);
}

// ---------------------------------------------------------------------------
// Kernel 2: cosine-distance GEMM via V_WMMA_F32_16X16X4_F32.
// Grid (NQ/8, NS, B), block 256 (8 waves). Each wave owns one (b,q,s) tile:
// one 16x16 dist matrix. K streamed in chunks of 64 through LDS.
// A(16x4 f32) frag: lanes 0-15 = rows, {v0,v1}={K,K+1}; lanes 16-31 {K+2,K+3}.
// B(4x16 f32) frag mirrors it with lanes = columns. LDS row stride 68 floats
// keeps the b64 fragment reads bank-conflict-free.
// ---------------------------------------------------------------------------
#define KC  64
#define LDA 68

__global__ __launch_bounds__(256) void otam_dist_kernel(
    const float* __restrict__ supp, const float* __restrict__ query,
    const float* __restrict__ norm, float* __restrict__ dist) {
  __shared__ float As[T_ * LDA];          // 4.25 KB, shared by all 8 waves
  __shared__ float Bs[8][T_ * LDA];       // 34 KB, one 16xKC tile per wave

  const int w    = threadIdx.x >> 5;
  const int lane = threadIdx.x & 31;
  const int hi   = lane >> 4;             // lane half selects K offset
  const int mm   = lane & 15;             // A row / B col / C col
  const int q = blockIdx.x * 8 + w;
  const int s = blockIdx.y;
  const int b = blockIdx.z;

  const float* sp = supp  + (size_t)((b * NS_ + s) * T_) * D_;
  const float* qp = query + (size_t)((b * NQ_ + q) * T_) * D_;

  v8f acc = {};

  for (int k0 = 0; k0 < D_; k0 += KC) {
    // Stage A tile: 256 threads x 1 float4 = 16x64.
    {
      int idx = threadIdx.x * 4;
      int row = idx >> 6, col = idx & 63;
      float4 v = *(const float4*)(sp + (size_t)row * D_ + k0 + col);
      float* d = &As[row * LDA + col];
      d[0] = v.x; d[1] = v.y; d[2] = v.z; d[3] = v.w;
    }
    // Stage this wave's B tile: 32 lanes x 8 float4 = 16x64.
#pragma unroll
    for (int j = 0; j < 8; ++j) {
      int e = j * 128 + lane * 4;
      int row = e >> 6, col = e & 63;
      float4 v = *(const float4*)(qp + (size_t)row * D_ + k0 + col);
      float* d = &Bs[w][row * LDA + col];
      d[0] = v.x; d[1] = v.y; d[2] = v.z; d[3] = v.w;
    }
    __syncthreads();

#if HAVE_WMMA_F32X4
#pragma unroll
    for (int j = 0; j < 16; ++j) {
      int c = 4 * j + 2 * hi;
      v2f av = *(const v2f*)(&As[mm * LDA + c]);
      v2f bv = *(const v2f*)(&Bs[w][mm * LDA + c]);
      acc = __builtin_amdgcn_wmma_f32_16x16x4_f32(
          false, av, false, bv, (short)0, acc, false, false);
    }
#else
#pragma unroll
    for (int r = 0; r < 8; ++r) {
      int l = r + 8 * hi;
      float sum = 0.f;
      for (int k = 0; k < KC; ++k)
        sum = fmaf(As[l * LDA + k], Bs[w][mm * LDA + k], sum);
      acc[r] += sum;
    }
#endif
    __syncthreads();
  }

  // Epilogue: dist = 1 - dots / max(sn*qn, eps); C layout: VGPR r -> row
  // l = r + 8*hi, lane&15 -> col m.
  const int n = (b * NQ_ + q) * NS_ + s;
  const float qn = norm[N_SUPP_ROWS + (b * NQ_ + q) * T_ + mm];
#pragma unroll
  for (int r = 0; r < 8; ++r) {
    int l = r + 8 * hi;
    float sn = norm[(b * NS_ + s) * T_ + l];
    float dv = 1.f - acc[r] / fmaxf(sn * qn, EPS_);
    dist[(size_t)n * (T_ * T_) + l * T_ + mm] = dv;
  }
}

// ---------------------------------------------------------------------------
// Kernel 3: relaxed OTAM band DP (soft-min). One lane per (matrix, dir).
// ---------------------------------------------------------------------------
__device__ __forceinline__ float lax(float x, float y) {  // logaddexp
  float mx = fmaxf(x, y);
  float mn = fminf(x, y);
  return mx + log1pf(__expf(mn - mx));
}

__global__ __launch_bounds__(256) void otam_dtw_kernel(
    const float* __restrict__ dist, float* __restrict__ tam) {
  int tid = blockIdx.x * blockDim.x + threadIdx.x;
  if (tid >= 2 * N_MAT) return;
  const int n = tid >> 1;
  const int dir = tid & 1;                  // 0 = forward, 1 = flipped
  const float* M = dist + (size_t)n * (T_ * T_);

  float prev[T_ + 2];
  float drow[T_];

  // Row 0: cumulative sum of dist_pad[0,:]
  {
    const float* rp = M + (dir ? 15 * T_ : 0);
#pragma unroll
    for (int i = 0; i < T_; ++i) drow[i] = rp[dir ? 15 - i : i];
    prev[0] = 0.f;
    float acc = 0.f;
#pragma unroll
    for (int m = 1; m <= T_; ++m) { acc += drow[m - 1]; prev[m] = acc; }
    prev[T_ + 1] = acc;                     // pad column adds 0
  }

#pragma unroll 1
  for (int l = 1; l < T_; ++l) {
    const float* rp = M + (dir ? (15 - l) * T_ : l * T_);
#pragma unroll
    for (int i = 0; i < T_; ++i) drow[i] = rp[dir ? 15 - i : i];
    float cur[T_ + 2];
    cur[0] = 0.f;
    float left = 0.f;
#pragma unroll
    for (int m = 1; m <= T_ + 1; ++m) {
      float d = (m <= T_) ? drow[m - 1] : 0.f;
      float x = lax(-prev[m - 1] * INVLAM_, -left * INVLAM_);   // {diag,left}
      if (m == 1 || m == T_ + 1)
        x = lax(x, -prev[m] * INVLAM_);                         // + {up}
      float val = -LAM_ * x + d;
      cur[m] = val;
      left = val;
    }
#pragma unroll
    for (int m = 0; m < T_ + 2; ++m) prev[m] = cur[m];
  }
  tam[dir * N_MAT + n] = prev[T_ + 1];
}

// ---------------------------------------------------------------------------
// Kernel 4: cross-entropy over (1024, 5) logits for both directions,
// tam averaging, and the final scalar loss. Single 256-thread block.
// out[0] = loss; out[1..5120] = 0.5*(tam1+tam2).
// ---------------------------------------------------------------------------
__global__ __launch_bounds__(256) void otam_reduce_kernel(
    const float* __restrict__ tam, const int* __restrict__ ys,
    float* __restrict__ out) {
  __shared__ float red[256];
  const int t = threadIdx.x;
  const float* t1 = tam;
  const float* t2 = tam + N_MAT;
  float lsum = 0.f;

  for (int i = t; i < B_ * NQ_; i += 256) {
    float a[NS_], c[NS_];
#pragma unroll
    for (int s = 0; s < NS_; ++s) {
      a[s] = t1[i * NS_ + s];
      c[s] = t2[i * NS_ + s];
    }
    int y = ys[i];
    // CE(-a, y) = a[y] + logsumexp(-a)
    float m1 = -a[0], m2 = -c[0];
#pragma unroll
    for (int s = 1; s < NS_; ++s) { m1 = fmaxf(m1, -a[s]); m2 = fmaxf(m2, -c[s]); }
    float e1 = 0.f, e2 = 0.f, ay = 0.f, cy = 0.f;
#pragma unroll
    for (int s = 0; s < NS_; ++s) {
      e1 += __expf(-a[s] - m1);
      e2 += __expf(-c[s] - m2);
      ay = (s == y) ? a[s] : ay;
      cy = (s == y) ? c[s] : cy;
      out[1 + i * NS_ + s] = 0.5f * (a[s] + c[s]);
    }
    lsum += (ay + m1 + logf(e1)) + (cy + m2 + logf(e2));
  }
  red[t] = lsum;
  __syncthreads();
  for (int st = 128; st >= 1; st >>= 1) {
    if (t < st) red[t] += red[t + st];
    __syncthreads();
  }
  if (t == 0) out[0] = red[0] / (2.f * (float)(B_ * NQ_));
}

// ---------------------------------------------------------------------------
extern "C" void kernel_launch(void* const* d_in, const int* in_sizes, int n_in,
                              void* d_out, int out_size, void* d_ws,
                              size_t ws_size, hipStream_t stream) {
  (void)in_sizes; (void)n_in; (void)out_size; (void)ws_size;
  const float* supp  = (const float*)d_in[0];
  const float* query = (const float*)d_in[1];
  const int*   ys    = (const int*)d_in[2];
  float* out  = (float*)d_out;
  float* ws   = (float*)d_ws;
  float* dist = ws + OFF_DIST;
  float* norm = ws + OFF_NORM;
  float* tam  = ws + OFF_TAM;

  // 1) row norms: 17024 rows, 8 waves/block
  otam_norms_kernel<<<dim3((N_SUPP_ROWS + N_QUERY_ROWS) / 8), dim3(256), 0,
                      stream>>>(supp, query, norm);
  // 2) cosine-distance GEMM: one 16x16 WMMA tile per (b,q,s)
  otam_dist_kernel<<<dim3(NQ_ / 8, NS_, B_), dim3(256), 0, stream>>>(
      supp, query, norm, dist);
  // 3) soft-min DP: one lane per (matrix, direction)
  otam_dtw_kernel<<<dim3((2 * N_MAT) / 256), dim3(256), 0, stream>>>(dist, tam);
  // 4) CE + tam average + loss
  otam_reduce_kernel<<<dim3(1), dim3(256), 0, stream>>>(tam, ys, out);
}